// GATEncoder_65103114273313
// MI455X (gfx1250) — compile-verified
//
#include <hip/hip_runtime.h>
#include <hip/hip_bf16.h>

// ---------------------------------------------------------------------------
// GAT encoder (3x GATConv, PyG semantics) for MI455X / gfx1250.
// GEMMs use V_WMMA_F32_16X16X4_F32 (exact f32 path). Edge phase is
// L2-resident atomic scatter/gather (wave32-aware: one wave per edge).
// ---------------------------------------------------------------------------

#define SLOPE 0.2f
#define FDIM 128          // in/out feature dim of every layer (HEADS*HID = EMB)

// enc_f32(-inf) = ~bits(-inf) = ~0xFF800000 = 0x007FFFFF
#define ENC_NEG_INF 0x007FFFFFu

typedef __attribute__((ext_vector_type(2))) float v2f;
typedef __attribute__((ext_vector_type(8))) float v8f;

// Device-pass-only sanity check (AMDGPU builtins are not declared in the
// host compilation pass).
#if defined(__HIP_DEVICE_COMPILE__)
#if !__has_builtin(__builtin_amdgcn_wmma_f32_16x16x4_f32)
#error "gfx1250 f32 WMMA builtin (__builtin_amdgcn_wmma_f32_16x16x4_f32) not available"
#endif
#endif

__device__ __forceinline__ v8f wmma_f32_16x16x4(v2f a, v2f b, v8f c) {
#if defined(__HIP_DEVICE_COMPILE__)
  // 8 args: (neg_a, A, neg_b, B, c_mod, C, reuse_a, reuse_b)
  return __builtin_amdgcn_wmma_f32_16x16x4_f32(false, a, false, b,
                                               (short)0, c, false, false);
#else
  (void)a; (void)b;
  return c;  // host-pass stub; never executed
#endif
}

// Order-preserving float <-> uint mapping for atomicMax-based segment_max.
__device__ __forceinline__ unsigned enc_f32(float f) {
  unsigned u = __float_as_uint(f);
  return (u & 0x80000000u) ? ~u : (u | 0x80000000u);
}
__device__ __forceinline__ float dec_f32(unsigned v) {
  return (v & 0x80000000u) ? __uint_as_float(v ^ 0x80000000u)
                           : __uint_as_float(~v);
}

// ---------------------------------------------------------------------------
// GEMM: H[N,128] = A[N,128] * W[128,128], f32 WMMA 16x16x4.
// Block = 256 threads (8 waves). Block handles 16 rows; wave w handles
// columns [16w, 16w+16). A-tile staged in LDS (stride 132 => conflict free).
// Requires N % 16 == 0 (N = 100000 = 6250*16).
// ---------------------------------------------------------------------------
__global__ __launch_bounds__(256) void gemm128_wmma(
    const float* __restrict__ A, const float* __restrict__ W,
    float* __restrict__ H) {
  __shared__ float As[16 * 132];
  const int tid = threadIdx.x;
  const int rowBase = blockIdx.x * 16;

  // Cooperative coalesced load of the 16x128 A tile (512 float4's).
  for (int i = tid; i < 512; i += 256) {
    const int r  = i >> 5;            // 32 float4 per row
    const int c4 = (i & 31) << 2;
    const float4 v = *(const float4*)(A + (size_t)(rowBase + r) * FDIM + c4);
    float* d = &As[r * 132 + c4];
    d[0] = v.x; d[1] = v.y; d[2] = v.z; d[3] = v.w;
  }
  __syncthreads();

  const int wave = tid >> 5;
  const int lane = tid & 31;
  const int m    = lane & 15;         // A row / C column-lane within half
  const int hi   = lane >> 4;         // 0: lanes 0-15, 1: lanes 16-31
  const int col  = wave * 16 + m;     // B / C column

  v8f c = {};
  for (int k = 0; k < FDIM; k += 4) {
    const int kk = k + 2 * hi;        // ISA 32-bit A/B operand layout
    v2f a, b;
    a.x = As[m * 132 + kk];
    a.y = As[m * 132 + kk + 1];
    b.x = W[(size_t)kk * FDIM + col];
    b.y = W[(size_t)(kk + 1) * FDIM + col];
    c = wmma_f32_16x16x4(a, b, c);
  }

  // C layout: VGPR r -> M = r (lanes 0-15) / M = r+8 (lanes 16-31).
  const int rowOff = rowBase + hi * 8;
#pragma unroll
  for (int r = 0; r < 8; ++r)
    H[(size_t)(rowOff + r) * FDIM + col] = c[r];
}

// ---------------------------------------------------------------------------
// Per-node attention scores: a_s[n,h] = <h[n,h,:], att_src[h,:]>, same for dst.
// ---------------------------------------------------------------------------
__global__ void attn_scores(const float* __restrict__ Hf,
                            const float* __restrict__ att_s,
                            const float* __restrict__ att_d,
                            float* __restrict__ a_s, float* __restrict__ a_d,
                            int n_nodes, int heads, int C) {
  const int idx = blockIdx.x * blockDim.x + threadIdx.x;
  if (idx >= n_nodes * heads) return;
  const int n = idx / heads;
  const int h = idx - n * heads;
  const float* row = Hf + (size_t)n * FDIM + h * C;
  const float* ws  = att_s + h * C;
  const float* wd  = att_d + h * C;
  float ss = 0.f, sd = 0.f;
  for (int c = 0; c < C; ++c) {
    const float v = row[c];
    ss += v * ws[c];
    sd += v * wd[c];
  }
  a_s[idx] = ss;
  a_d[idx] = sd;
}

__global__ void fill_u32(unsigned* __restrict__ p, unsigned v, int n) {
  const int i = blockIdx.x * blockDim.x + threadIdx.x;
  if (i < n) p[i] = v;
}

// Edges are [0,E) from edge_index, [E, E+N) are implicit self-loops.
__device__ __forceinline__ void edge_endpoints(const int* __restrict__ ei,
                                               int e, int E, int& s, int& d) {
  if (e < E) { s = ei[e]; d = ei[E + e]; }
  else       { s = d = e - E; }
}

__global__ void edge_max(const int* __restrict__ ei,
                         const float* __restrict__ a_s,
                         const float* __restrict__ a_d,
                         unsigned* __restrict__ m_enc,
                         int E, int EE, int heads) {
  const int e = blockIdx.x * blockDim.x + threadIdx.x;
  if (e >= EE) return;
  int s, d; edge_endpoints(ei, e, E, s, d);
  for (int h = 0; h < heads; ++h) {
    float v = a_s[s * heads + h] + a_d[d * heads + h];
    v = (v > 0.f) ? v : SLOPE * v;                 // leaky_relu
    atomicMax(&m_enc[d * heads + h], enc_f32(v));
  }
}

__global__ void edge_denom(const int* __restrict__ ei,
                           const float* __restrict__ a_s,
                           const float* __restrict__ a_d,
                           const unsigned* __restrict__ m_enc,
                           float* __restrict__ denom,
                           int E, int EE, int heads) {
  const int e = blockIdx.x * blockDim.x + threadIdx.x;
  if (e >= EE) return;
  int s, d; edge_endpoints(ei, e, E, s, d);
  for (int h = 0; h < heads; ++h) {
    float v = a_s[s * heads + h] + a_d[d * heads + h];
    v = (v > 0.f) ? v : SLOPE * v;
    const float m = dec_f32(m_enc[d * heads + h]);
    atomicAdd(&denom[d * heads + h], expf(v - m));
  }
}

// One wave32 per edge; lane handles 4 channels: float4 gather of h[src],
// scale by alpha, 4x global f32 atomic adds into out[dst].
__global__ __launch_bounds__(256) void edge_aggregate(
    const int* __restrict__ ei, const float* __restrict__ Hf,
    const float* __restrict__ a_s, const float* __restrict__ a_d,
    const unsigned* __restrict__ m_enc, const float* __restrict__ denom,
    float* __restrict__ acc, int E, int EE, int heads, int C) {
  const int e = blockIdx.x * 8 + (threadIdx.x >> 5);
  if (e >= EE) return;
  const int lane = threadIdx.x & 31;
  int s, d; edge_endpoints(ei, e, E, s, d);

  const int ch = lane * 4;
  const int h  = ch / C;                 // C=32 -> lane>>3 ; C=128 -> 0
  float v = a_s[s * heads + h] + a_d[d * heads + h];
  v = (v > 0.f) ? v : SLOPE * v;
  const float m = dec_f32(m_enc[d * heads + h]);
  const float alpha = expf(v - m) / denom[d * heads + h];

  const float4 hv = *(const float4*)(Hf + (size_t)s * FDIM + ch);
  float* o = acc + (size_t)d * FDIM + ch;
  atomicAdd(o + 0, hv.x * alpha);
  atomicAdd(o + 1, hv.y * alpha);
  atomicAdd(o + 2, hv.z * alpha);
  atomicAdd(o + 3, hv.w * alpha);
}

__global__ void bias_act(float* __restrict__ out, const float* __restrict__ b,
                         int n, int do_elu) {
  const int i = blockIdx.x * blockDim.x + threadIdx.x;
  if (i >= n) return;
  float v = out[i] + b[i & (FDIM - 1)];
  if (do_elu) v = (v > 0.f) ? v : (expf(v) - 1.f);
  out[i] = v;
}

// ---------------------------------------------------------------------------
// Host side
// ---------------------------------------------------------------------------
static void run_layer(const float* in, const int* ei, int E, int EE, int n_nodes,
                      const float* W, const float* att_s, const float* att_d,
                      const float* b, int heads, int C, int do_elu,
                      float* Hf, float* a_s, float* a_d, unsigned* m_enc,
                      float* denom, float* acc, hipStream_t stream) {
  const int BS = 256;
  const int nh = n_nodes * heads;
  const int nf = n_nodes * FDIM;

  gemm128_wmma<<<n_nodes / 16, BS, 0, stream>>>(in, W, Hf);
  attn_scores<<<(nh + BS - 1) / BS, BS, 0, stream>>>(Hf, att_s, att_d,
                                                     a_s, a_d, n_nodes, heads, C);
  fill_u32<<<(nh + BS - 1) / BS, BS, 0, stream>>>(m_enc, ENC_NEG_INF, nh);
  fill_u32<<<(nh + BS - 1) / BS, BS, 0, stream>>>((unsigned*)denom, 0u, nh);
  fill_u32<<<(nf + BS - 1) / BS, BS, 0, stream>>>((unsigned*)acc, 0u, nf);
  edge_max<<<(EE + BS - 1) / BS, BS, 0, stream>>>(ei, a_s, a_d, m_enc, E, EE, heads);
  edge_denom<<<(EE + BS - 1) / BS, BS, 0, stream>>>(ei, a_s, a_d, m_enc, denom,
                                                    E, EE, heads);
  edge_aggregate<<<(EE + 7) / 8, BS, 0, stream>>>(ei, Hf, a_s, a_d, m_enc, denom,
                                                  acc, E, EE, heads, C);
  bias_act<<<(nf + BS - 1) / BS, BS, 0, stream>>>(acc, b, nf, do_elu);
}

extern "C" void kernel_launch(void* const* d_in, const int* in_sizes, int n_in,
                              void* d_out, int out_size, void* d_ws, size_t ws_size,
                              hipStream_t stream) {
  const float* x   = (const float*)d_in[0];
  const int*   ei  = (const int*)d_in[1];
  const float* W1  = (const float*)d_in[2];
  const float* as1 = (const float*)d_in[3];
  const float* ad1 = (const float*)d_in[4];
  const float* b1  = (const float*)d_in[5];
  const float* W2  = (const float*)d_in[6];
  const float* as2 = (const float*)d_in[7];
  const float* ad2 = (const float*)d_in[8];
  const float* b2  = (const float*)d_in[9];
  const float* W3  = (const float*)d_in[10];
  const float* as3 = (const float*)d_in[11];
  const float* ad3 = (const float*)d_in[12];
  const float* b3  = (const float*)d_in[13];

  const int n_nodes = in_sizes[0] / FDIM;   // 100000
  const int E       = in_sizes[1] / 2;      // 1600000
  const int EE      = E + n_nodes;          // + self loops

  // Workspace carve-up.
  char* w = (char*)d_ws;
  float* B0 = (float*)w; w += (size_t)n_nodes * FDIM * sizeof(float);   // features
  float* B1 = (float*)w; w += (size_t)n_nodes * FDIM * sizeof(float);   // act/accum
  float*    a_s   = (float*)w;    w += (size_t)n_nodes * 4 * sizeof(float);
  float*    a_d   = (float*)w;    w += (size_t)n_nodes * 4 * sizeof(float);
  unsigned* m_enc = (unsigned*)w; w += (size_t)n_nodes * 4 * sizeof(unsigned);
  float*    denom = (float*)w;    w += (size_t)n_nodes * 4 * sizeof(float);

  float* out = (float*)d_out;

  // Layer 1: x -> B1 (elu applied in-place)
  run_layer(x,  ei, E, EE, n_nodes, W1, as1, ad1, b1, 4, 32, 1,
            B0, a_s, a_d, m_enc, denom, B1, stream);
  // Layer 2: features in B0; accumulator reuses B1 after the GEMM has read it
  // (stream ordering guarantees the GEMM completes before fill/aggregate).
  run_layer(B1, ei, E, EE, n_nodes, W2, as2, ad2, b2, 4, 32, 1,
            B0, a_s, a_d, m_enc, denom, B1, stream);
  // Layer 3 (heads=1, C=128): accumulate directly into d_out, bias only.
  run_layer(B1, ei, E, EE, n_nodes, W3, as3, ad3, b3, 1, 128, 0,
            B0, a_s, a_d, m_enc, denom, out, stream);
}